// LearnedRotation_78658031059523
// MI455X (gfx1250) — compile-verified
//
#include <hip/hip_runtime.h>
#include <hip/hip_bf16.h>

// ---------------------------------------------------------------------------
// LearnedRotation: out = sum_i softmax(mean_HW(x) @ W_sel + b_sel)[b,i] * rot90(x, i)
// B=32, H=W=128, C=128, fp32.  Memory-bound: ~768MB min traffic @ 23.3TB/s.
// ---------------------------------------------------------------------------

#define B_  32
#define HW_ 128
#define C_  128
#define SPATIAL (HW_ * HW_)        // 16384
#define SLICES_PER_B 64            // 256 spatial positions per slice

typedef __attribute__((ext_vector_type(2))) float v2f;
typedef __attribute__((ext_vector_type(4))) float v4f;
typedef __attribute__((ext_vector_type(8))) float v8f;

// ---------------------------------------------------------------------------
// Kernel 1: partial spatial sums.  grid = B*64 blocks, 256 threads.
// Each block sums 256 spatial positions (all 128 channels) of one batch image
// into partial[b][slice][c].  Fixed-order LDS tree => deterministic.
// Default (RT) loads: x is re-read by rotate_combine, let L2 (192MB) hold it.
// ---------------------------------------------------------------------------
__global__ __launch_bounds__(256) void pool_partial(const float* __restrict__ x,
                                                    float* __restrict__ partial) {
    const int blk   = blockIdx.x;       // b*64 + slice
    const int b     = blk >> 6;
    const int slice = blk & 63;
    const int t     = threadIdx.x;
    const int c4    = t & 31;           // float4 channel group (32 * 4 = 128 ch)
    const int sub   = t >> 5;           // 0..7 sub-position

    const float4* x4 = (const float4*)x;
    // float4 index of x[b, s, c4*4]:  (b*16384 + s)*32 + c4
    long base = ((long)b * SPATIAL + slice * 256 + sub) * 32 + c4;

    float4 acc = make_float4(0.f, 0.f, 0.f, 0.f);
    #pragma unroll 4
    for (int j = 0; j < 32; ++j) {
        float4 v = x4[base + (long)j * 8 * 32];   // positions sub + 8*j
        acc.x += v.x; acc.y += v.y; acc.z += v.z; acc.w += v.w;
    }

    __shared__ float4 red[8][32];
    red[sub][c4] = acc;
    __syncthreads();
    for (int off = 4; off >= 1; off >>= 1) {
        if (sub < off) {
            float4 a = red[sub][c4];
            float4 c = red[sub + off][c4];
            a.x += c.x; a.y += c.y; a.z += c.z; a.w += c.w;
            red[sub][c4] = a;
        }
        __syncthreads();
    }
    if (sub == 0) {
        float4* p4 = (float4*)partial;
        p4[(long)blk * 32 + c4] = red[0][c4];     // partial[b][slice][c..c+3]
    }
}

// ---------------------------------------------------------------------------
// Kernel 2: reduce 64 slice partials -> pooled[b][c] (sum, not yet mean).
// 4096 (b,c) pairs; serial fixed-order loop => deterministic.
// ---------------------------------------------------------------------------
__global__ __launch_bounds__(256) void pool_reduce(const float* __restrict__ partial,
                                                   float* __restrict__ pooled) {
    const int i = blockIdx.x * 256 + threadIdx.x;   // (b*128 + c), 0..4095
    if (i >= B_ * C_) return;
    const int b = i >> 7;
    const int c = i & 127;
    float s = 0.f;
    #pragma unroll 8
    for (int sl = 0; sl < SLICES_PER_B; ++sl)
        s += partial[(b * SLICES_PER_B + sl) * C_ + c];
    pooled[i] = s;
}

// ---------------------------------------------------------------------------
// Kernel 3: selector GEMM + softmax on ONE wave32 using V_WMMA_F32_16X16X4_F32.
//   logits[32,4] = (pooled/16384)[32,128] @ W_sel[128,4] + b_sel
// Two 16-row D tiles (M=32), K=128 in 32 chunks of 4, N padded 4->16 w/ zeros.
// A 16x4 layout: lane L, vgpr v -> m=L%16, k=(L/16)*2+v
// B 4x16 layout: lane L, vgpr v -> n=L%16, k=(L/16)*2+v
// D 16x16 layout: lane L, vgpr v -> n=L%16, m=(L/16)*8+v
// B-fragment padding is branch-free: unconditional in-bounds load of
// W_sel[k][n&3] scaled by a 0/1 lane mask (keeps EXEC all-ones, no saveexec).
// ---------------------------------------------------------------------------
__global__ __launch_bounds__(32) void selector_wmma(const float* __restrict__ pooled,
                                                    const float* __restrict__ W_sel,
                                                    const float* __restrict__ b_sel,
                                                    float* __restrict__ weights) {
    const int lane = threadIdx.x;
    const int m    = lane & 15;          // row for A, col for B/D
    const int n    = lane & 15;
    const int nm   = n & 3;              // clamped (always in-bounds) B column
    const float bmask = (n < 4) ? 1.0f : 0.0f;
    const int kg   = (lane >> 4) << 1;   // 0 or 2: k sub-offset within chunk
    const float inv = 1.0f / (float)SPATIAL;

    v8f d0 = {0.f, 0.f, 0.f, 0.f, 0.f, 0.f, 0.f, 0.f};   // rows 0..15
    v8f d1 = {0.f, 0.f, 0.f, 0.f, 0.f, 0.f, 0.f, 0.f};   // rows 16..31

    for (int kc = 0; kc < C_; kc += 4) {
        const int k0 = kc + kg;
        v2f a0, a1, bf;
        a0.x = pooled[m * C_ + k0]            * inv;
        a0.y = pooled[m * C_ + k0 + 1]        * inv;
        a1.x = pooled[(m + 16) * C_ + k0]     * inv;
        a1.y = pooled[(m + 16) * C_ + k0 + 1] * inv;
        bf.x = W_sel[k0 * 4 + nm]       * bmask;
        bf.y = W_sel[(k0 + 1) * 4 + nm] * bmask;
        d0 = __builtin_amdgcn_wmma_f32_16x16x4_f32(false, a0, false, bf,
                                                   (short)0, d0, false, false);
        d1 = __builtin_amdgcn_wmma_f32_16x16x4_f32(false, a1, false, bf,
                                                   (short)0, d1, false, false);
    }

    // Scatter the 4 valid columns of each row to LDS, then softmax per row.
    __shared__ float logits[32][4];
    const int grp = lane >> 4;
    if (n < 4) {
        #pragma unroll
        for (int v = 0; v < 8; ++v) {
            logits[grp * 8 + v][n]      = d0[v];
            logits[16 + grp * 8 + v][n] = d1[v];
        }
    }
    __syncthreads();

    // one batch row per lane
    float l[4];
    #pragma unroll
    for (int i = 0; i < 4; ++i) l[i] = logits[lane][i] + b_sel[i];
    float mx = fmaxf(fmaxf(l[0], l[1]), fmaxf(l[2], l[3]));
    float e[4], s = 0.f;
    #pragma unroll
    for (int i = 0; i < 4; ++i) { e[i] = __expf(l[i] - mx); s += e[i]; }
    float rs = 1.0f / s;
    #pragma unroll
    for (int i = 0; i < 4; ++i) weights[lane * 4 + i] = e[i] * rs;
}

// ---------------------------------------------------------------------------
// Kernel 4: fused 4-rotation combine over rot90 orbits.
// Orbit rep: h,w in [0,64). One thread loads x at the 4 orbit points
// (float4 of channels) and writes all 4 outputs => 1x read + 1x write total.
// Output is write-only, never re-read: non-temporal stores keep the 256MB
// out-stream from evicting x from the 192MB L2 (x may still be resident from
// pool_partial when we gather here).  NT store goes through a clang
// ext_vector float4 (same 16B layout) since __builtin_nontemporal_store
// rejects HIP_vector_type structs.
// grid = B * 512 blocks (512 = 64*64*32/256), 256 threads.
// ---------------------------------------------------------------------------
__device__ __forceinline__ float4 f4fma(float s, float4 v, float4 a) {
    a.x = fmaf(s, v.x, a.x); a.y = fmaf(s, v.y, a.y);
    a.z = fmaf(s, v.z, a.z); a.w = fmaf(s, v.w, a.w);
    return a;
}
__device__ __forceinline__ float4 f4mul(float s, float4 v) {
    return make_float4(s * v.x, s * v.y, s * v.z, s * v.w);
}
__device__ __forceinline__ void nt_store4(float4* p, float4 v) {
    v4f t;
    t.x = v.x; t.y = v.y; t.z = v.z; t.w = v.w;
    __builtin_nontemporal_store(t, (v4f*)p);
}

__global__ __launch_bounds__(256) void rotate_combine(const float* __restrict__ x,
                                                      const float* __restrict__ weights,
                                                      float* __restrict__ out) {
    const int b     = blockIdx.x >> 9;          // 512 blocks per batch image
    const int inner = blockIdx.x & 511;
    const int t     = inner * 256 + threadIdx.x; // 0..131071
    const int c4    = t & 31;                    // float4 channel group
    const int r     = t >> 5;
    const int w     = r & 63;                    // quadrant coords
    const int h     = r >> 6;

    const float w0 = weights[b * 4 + 0];
    const float w1 = weights[b * 4 + 1];
    const float w2 = weights[b * 4 + 2];
    const float w3 = weights[b * 4 + 3];

    const float4* x4 = (const float4*)x;
    float4*       o4 = (float4*)out;

    // float4 index of x[b, hh, ww, c4*4]
    const long bb = (long)b * HW_;
    #define IDX(hh, ww) (((bb + (hh)) * HW_ + (ww)) * 32 + c4)
    const long p0 = IDX(h, w);
    const long p1 = IDX(w, 127 - h);
    const long p2 = IDX(127 - h, 127 - w);
    const long p3 = IDX(127 - w, h);
    #undef IDX

    const float4 x0 = x4[p0];
    const float4 x1 = x4[p1];
    const float4 x2 = x4[p2];
    const float4 x3 = x4[p3];

    // out[p_i] = w0*x_i + w1*x_{i+1} + w2*x_{i+2} + w3*x_{i+3}   (cyclic)
    nt_store4(&o4[p0], f4fma(w3, x3, f4fma(w2, x2, f4fma(w1, x1, f4mul(w0, x0)))));
    nt_store4(&o4[p1], f4fma(w3, x0, f4fma(w2, x3, f4fma(w1, x2, f4mul(w0, x1)))));
    nt_store4(&o4[p2], f4fma(w3, x1, f4fma(w2, x0, f4fma(w1, x3, f4mul(w0, x2)))));
    nt_store4(&o4[p3], f4fma(w3, x2, f4fma(w2, x1, f4fma(w1, x0, f4mul(w0, x3)))));
}

// ---------------------------------------------------------------------------
// Host-side launcher.  d_ws layout (floats):
//   [0 .. 262143]           partial sums  (32 * 64 * 128)   = 1 MB
//   [262144 .. 266239]      pooled        (32 * 128)
//   [266240 .. 266367]      weights       (32 * 4)
// ---------------------------------------------------------------------------
extern "C" void kernel_launch(void* const* d_in, const int* in_sizes, int n_in,
                              void* d_out, int out_size, void* d_ws, size_t ws_size,
                              hipStream_t stream) {
    const float* x     = (const float*)d_in[0];
    const float* W_sel = (const float*)d_in[1];
    const float* b_sel = (const float*)d_in[2];
    float*       out   = (float*)d_out;

    float* partial = (float*)d_ws;
    float* pooled  = partial + (B_ * SLICES_PER_B * C_);
    float* weights = pooled + (B_ * C_);

    pool_partial  <<<B_ * SLICES_PER_B, 256, 0, stream>>>(x, partial);
    pool_reduce   <<<(B_ * C_ + 255) / 256, 256, 0, stream>>>(partial, pooled);
    selector_wmma <<<1, 32, 0, stream>>>(pooled, W_sel, b_sel, weights);
    rotate_combine<<<B_ * 512, 256, 0, stream>>>(x, weights, out);
}